// InterRegion_59098749993442
// MI455X (gfx1250) — compile-verified
//
#include <hip/hip_runtime.h>

#define NB 2
#define NC 512
#define NH 128
#define NW 256
#define NHW (NH*NW)        // 32768
#define NK 19
#define CBLK 64
#define NCB (NC/CBLK)      // 8 channel blocks
#define PCHUNK 1024        // pixels per workgroup chunk (256 thr * 4 px)
#define NPC (NHW/PCHUNK)   // 32 pixel chunks
#define THREADS 256
#define CSTRIDE 32         // LDS row stride, zero-padded classes 19..31

typedef float v2f __attribute__((ext_vector_type(2)));
typedef float v8f __attribute__((ext_vector_type(8)));

// ---------------- kernel 0: zero scratch ----------------
__global__ void ir_zero(float* __restrict__ ws, int n) {
  int i = blockIdx.x * blockDim.x + threadIdx.x;
  if (i < n) ws[i] = 0.f;
}

// ---------------- kernel 1: bandwidth-bound segment reduction ----------------
// sums[b][c][k] = sum over pixels with label k of feat[b][c][p]; counts[b][k].
// Coalesced float4 loads (4 px/lane), labels cached in registers across the
// whole channel loop, LDS ds_add_f32 scatter, global f32 atomics for partials.
__global__ __launch_bounds__(THREADS) void ir_scatter(
    const float* __restrict__ S, const float* __restrict__ T,
    const int* __restrict__ tgt,
    float* __restrict__ gS, float* __restrict__ gT, float* __restrict__ gcnt) {
  __shared__ float accS[CBLK * NK];
  __shared__ float accT[CBLK * NK];
  __shared__ float cnt[NK];

  const int tid = threadIdx.x;
  const int wg  = blockIdx.x;
  const int cb  = wg % NCB;
  const int pc  = (wg / NCB) % NPC;
  const int b   = wg / (NCB * NPC);

  for (int i = tid; i < CBLK * NK; i += THREADS) { accS[i] = 0.f; accT[i] = 0.f; }
  if (tid < NK) cnt[tid] = 0.f;
  __syncthreads();

  const int p0 = pc * PCHUNK + tid * 4;
  const int l0 = tgt[b * NHW + p0 + 0];
  const int l1 = tgt[b * NHW + p0 + 1];
  const int l2 = tgt[b * NHW + p0 + 2];
  const int l3 = tgt[b * NHW + p0 + 3];

  if (cb == 0) {
    atomicAdd(&cnt[l0], 1.f); atomicAdd(&cnt[l1], 1.f);
    atomicAdd(&cnt[l2], 1.f); atomicAdd(&cnt[l3], 1.f);
  }

  const int c0 = cb * CBLK;
  size_t base = ((size_t)b * NC + c0) * NHW + p0;
  #pragma unroll 4
  for (int cr = 0; cr < CBLK; ++cr) {
    const float4 vS = *reinterpret_cast<const float4*>(S + base);
    const float4 vT = *reinterpret_cast<const float4*>(T + base);
    float* aS = &accS[cr * NK];
    float* aT = &accT[cr * NK];
    atomicAdd(&aS[l0], vS.x); atomicAdd(&aS[l1], vS.y);
    atomicAdd(&aS[l2], vS.z); atomicAdd(&aS[l3], vS.w);
    atomicAdd(&aT[l0], vT.x); atomicAdd(&aT[l1], vT.y);
    atomicAdd(&aT[l2], vT.z); atomicAdd(&aT[l3], vT.w);
    base += NHW;
  }
  __syncthreads();

  const size_t gbase = ((size_t)b * NC + c0) * NK;
  for (int i = tid; i < CBLK * NK; i += THREADS) {
    atomicAdd(&gS[gbase + i], accS[i]);
    atomicAdd(&gT[gbase + i], accT[i]);
  }
  if (cb == 0 && tid < NK) atomicAdd(&gcnt[b * NK + tid], cnt[tid]);
}

// ---------------- kernel 2: centers -> WMMA Gram matrices -> cosine MSE ----
// dots[i][j] = sum_c center[c][i]*center[c][j]  (19x19 padded to 32x32),
// via V_WMMA_F32_16X16X4_F32 on wave 0. One workgroup total.
__global__ __launch_bounds__(256) void ir_finalize(
    const float* __restrict__ gS, const float* __restrict__ gT,
    const float* __restrict__ gcnt, float* __restrict__ out) {
  extern __shared__ float smem[];
  float* centS = smem;                       // NC * CSTRIDE (zero-padded)
  float* centT = centS + NC * CSTRIDE;       // NC * CSTRIDE
  float* dotsS = centT + NC * CSTRIDE;       // 32*32
  float* dotsT = dotsS + 32 * 32;            // 32*32
  float* nrmS  = dotsT + 32 * 32;            // 32
  float* nrmT  = nrmS + 32;                  // 32
  __shared__ float red[256];

  const int tid = threadIdx.x;
  const int lane = tid & 31;
  const int hf = lane >> 4;   // half-wave selects K=0/1 vs K=2/3 per ISA layout
  const int m  = lane & 15;   // M (A rows) / N (B cols) index within tile

  float total = 0.f;
  for (int b = 0; b < NB; ++b) {
    __syncthreads();
    // centers = sums / (counts + 1e-6); classes 19..31 zero-padded so the
    // WMMA operand fetch needs no bounds check (no exec-mask churn).
    for (int i = tid; i < NC * CSTRIDE; i += 256) {
      int c = i >> 5, k = i & 31;
      float vS = 0.f, vT = 0.f;
      if (k < NK) {
        float inv = 1.f / (gcnt[b * NK + k] + 1e-6f);
        vS = gS[((size_t)b * NC + c) * NK + k] * inv;
        vT = gT[((size_t)b * NC + c) * NK + k] * inv;
      }
      centS[i] = vS;
      centT[i] = vT;
    }
    __syncthreads();

    if (tid < 32) {  // wave 0, EXEC all ones
      v8f aS00 = {}, aS01 = {}, aS10 = {}, aS11 = {};
      v8f aT00 = {}, aT01 = {}, aT10 = {}, aT11 = {};
      for (int c0 = 0; c0 < NC; c0 += 4) {
        const int cA = c0 + 2 * hf;       // K=0 (hf=0) or K=2 (hf=1)
        const int cB = cA + 1;            // K=1 or K=3
        // operand tile 0: classes m (0..15); tile 1: classes 16+m (padded 0)
        v2f s0 = { centS[cA * CSTRIDE + m],      centS[cB * CSTRIDE + m]      };
        v2f s1 = { centS[cA * CSTRIDE + 16 + m], centS[cB * CSTRIDE + 16 + m] };
        v2f t0 = { centT[cA * CSTRIDE + m],      centT[cB * CSTRIDE + m]      };
        v2f t1 = { centT[cA * CSTRIDE + 16 + m], centT[cB * CSTRIDE + 16 + m] };
        // D = A x B + C ; matrices symmetric so A-tile == B-tile operand
        aS00 = __builtin_amdgcn_wmma_f32_16x16x4_f32(false, s0, false, s0, (short)0, aS00, false, false);
        aS01 = __builtin_amdgcn_wmma_f32_16x16x4_f32(false, s0, false, s1, (short)0, aS01, false, false);
        aS10 = __builtin_amdgcn_wmma_f32_16x16x4_f32(false, s1, false, s0, (short)0, aS10, false, false);
        aS11 = __builtin_amdgcn_wmma_f32_16x16x4_f32(false, s1, false, s1, (short)0, aS11, false, false);
        aT00 = __builtin_amdgcn_wmma_f32_16x16x4_f32(false, t0, false, t0, (short)0, aT00, false, false);
        aT01 = __builtin_amdgcn_wmma_f32_16x16x4_f32(false, t0, false, t1, (short)0, aT01, false, false);
        aT10 = __builtin_amdgcn_wmma_f32_16x16x4_f32(false, t1, false, t0, (short)0, aT10, false, false);
        aT11 = __builtin_amdgcn_wmma_f32_16x16x4_f32(false, t1, false, t1, (short)0, aT11, false, false);
      }
      // C/D layout: lane L, reg r -> M = r + 8*(L>=16), N = L&15
      #pragma unroll
      for (int r = 0; r < 8; ++r) {
        int row = r + 8 * hf;
        dotsS[row * 32 + m]             = aS00[r];
        dotsS[row * 32 + 16 + m]        = aS01[r];
        dotsS[(row + 16) * 32 + m]      = aS10[r];
        dotsS[(row + 16) * 32 + 16 + m] = aS11[r];
        dotsT[row * 32 + m]             = aT00[r];
        dotsT[row * 32 + 16 + m]        = aT01[r];
        dotsT[(row + 16) * 32 + m]      = aT10[r];
        dotsT[(row + 16) * 32 + 16 + m] = aT11[r];
      }
    }
    __syncthreads();

    if (tid < NK) {
      nrmS[tid] = fmaxf(sqrtf(dotsS[tid * 32 + tid]), 1e-8f);
      nrmT[tid] = fmaxf(sqrtf(dotsT[tid * 32 + tid]), 1e-8f);
    }
    __syncthreads();

    float part = 0.f;
    for (int i = tid; i < NK * NK; i += 256) {
      int r = i / NK, c = i - r * NK;
      float sS = dotsS[r * 32 + c] / (nrmS[r] * nrmS[c]);
      float sT = dotsT[r * 32 + c] / (nrmT[r] * nrmT[c]);
      float d = sS - sT;
      part += d * d;
    }
    red[tid] = part;
    __syncthreads();
    for (int s = 128; s > 0; s >>= 1) {
      if (tid < s) red[tid] += red[tid + s];
      __syncthreads();
    }
    total += red[0];
  }
  if (tid == 0) out[0] = total / (float)(NB * NK * NK);
}

extern "C" void kernel_launch(void* const* d_in, const int* in_sizes, int n_in,
                              void* d_out, int out_size, void* d_ws, size_t ws_size,
                              hipStream_t stream) {
  const float* S   = (const float*)d_in[0];
  const float* T   = (const float*)d_in[1];
  const int*   tgt = (const int*)d_in[2];

  float* ws = (float*)d_ws;
  float* gS = ws;                      // NB*NC*NK
  float* gT = gS + NB * NC * NK;       // NB*NC*NK
  float* gc = gT + NB * NC * NK;       // NB*NK

  const int nz = 2 * NB * NC * NK + NB * NK;
  hipLaunchKernelGGL(ir_zero, dim3((nz + 255) / 256), dim3(256), 0, stream, ws, nz);

  hipLaunchKernelGGL(ir_scatter, dim3(NB * NPC * NCB), dim3(THREADS), 0, stream,
                     S, T, tgt, gS, gT, gc);

  const size_t shmem = (size_t)(2 * NC * CSTRIDE + 2 * 32 * 32 + 64) * sizeof(float);
  hipLaunchKernelGGL(ir_finalize, dim3(1), dim3(256), shmem, stream,
                     gS, gT, gc, (float*)d_out);
}